// MultiscaleSidechainEncoder_26740466384956
// MI455X (gfx1250) — compile-verified
//
#include <hip/hip_runtime.h>

#define RR 1024
#define APR 14
#define AA 14336          // RR*APR
#define KK 30
#define CC 4
#define CHN 16
#define NH 32
#define ECH 64
#define EKNN (AA*KK)      // 430080
#define ERES (RR*KK)      // 30720

typedef float v2f __attribute__((ext_vector_type(2)));
typedef float v8f __attribute__((ext_vector_type(8)));

// ---------------------------------------------------------------------------
// Async global->LDS copy (CDNA5 GLOBAL_LOAD_ASYNC_TO_LDS_B128, ASYNCcnt).
// Each active lane moves 16B; one instruction moves 512B per wave32.
// LDS operand is the wave-relative 32-bit LDS byte address (= low 32 bits of
// the generic pointer, per ISA aperture rules).
// ---------------------------------------------------------------------------
__device__ __forceinline__ void async_ld16(uint32_t lds_off, uint64_t gaddr) {
  asm volatile("global_load_async_to_lds_b128 %0, %1, off"
               :: "v"(lds_off), "v"(gaddr)
               : "memory");
}

__device__ __forceinline__ void async_wait0() {
#if __has_builtin(__builtin_amdgcn_s_wait_asynccnt)
  __builtin_amdgcn_s_wait_asynccnt(0);
#else
  asm volatile("s_wait_asynccnt 0x0" ::: "memory");
#endif
}

// contiguous global -> contiguous LDS, nbytes multiple of 512, 16B aligned
__device__ __forceinline__ void async_copy_contig(void* lds, const void* g,
                                                  int nbytes) {
  const int l = threadIdx.x & 31;
  uint32_t l0 = (uint32_t)(uintptr_t)lds + (uint32_t)l * 16u;
  uint64_t g0 = (uint64_t)(uintptr_t)g + (uint64_t)l * 16u;
  for (int off = 0; off < nbytes; off += 512)
    async_ld16(l0 + (uint32_t)off, g0 + (uint64_t)off);
}

// 16 rows x 256B contiguous global tile -> LDS rows with 272B (68-float) stride
__device__ __forceinline__ void async_copy_tile_padded(void* lds, const void* g) {
  const int l = threadIdx.x & 31;
  const int chunk = l & 15;   // 16B chunk within a row
  const int rhalf = l >> 4;   // row parity
  for (int rp = 0; rp < 8; ++rp) {
    int row = rp * 2 + rhalf;
    uint32_t lo = (uint32_t)(uintptr_t)lds + (uint32_t)row * 272u +
                  (uint32_t)chunk * 16u;
    uint64_t go = (uint64_t)(uintptr_t)g + (uint64_t)row * 256u +
                  (uint64_t)chunk * 16u;
    async_ld16(lo, go);
  }
}

// ---------------------------------------------------------------------------
// fp32 WMMA 16x16xK tile GEMM, K multiple of 4.
// A: 16 rows x K floats at As (row stride lda)
// B: K x N floats at Bs (row stride ldb), output col tile [n0, n0+16)
// Lane layout per CDNA5 ISA (V_WMMA_F32_16X16X4_F32):
//   A: lane l -> row m=l&15 ; VGPR0/1 = K (2*(l>>4)) / (2*(l>>4)+1)
//   B: lane l -> col n=l&15 ; VGPR0/1 = K rows 2*(l>>4), 2*(l>>4)+1
//   D: VGPR j -> row j + 8*(l>>4), col l&15
// ---------------------------------------------------------------------------
__device__ __forceinline__ v8f gemm_tile(const float* As, int lda,
                                         const float* Bs, int ldb,
                                         int K, int n0, v8f acc) {
  const int lane = threadIdx.x & 31;
  const int n = lane & 15;
  const int kh = lane >> 4;
  const int m = n;
  for (int k0 = 0; k0 < K; k0 += 4) {
    v2f a, b;
    a.x = As[m * lda + k0 + 2 * kh];
    a.y = As[m * lda + k0 + 2 * kh + 1];
    b.x = Bs[(k0 + 2 * kh) * ldb + n0 + n];
    b.y = Bs[(k0 + 2 * kh + 1) * ldb + n0 + n];
    acc = __builtin_amdgcn_wmma_f32_16x16x4_f32(false, a, false, b,
                                                (short)0, acc, false, false);
  }
  return acc;
}

__device__ __forceinline__ v8f vzero8() {
  v8f z = {0.f,0.f,0.f,0.f,0.f,0.f,0.f,0.f};
  return z;
}

// ---------------------------------------------------------------------------
// q/k/v precompute: q=x[:,0]@Wq, k=x[:,0]@Wk, v[c]=x[:,c]@Wv.  16 nodes/block.
// ---------------------------------------------------------------------------
__global__ void k_qkv(const float* __restrict__ x,
                      const float* __restrict__ Wq, const float* __restrict__ Wk,
                      const float* __restrict__ Wv,
                      float* __restrict__ q, float* __restrict__ kk,
                      float* __restrict__ v) {
  __shared__ __align__(16) float Ax[16 * 68];
  __shared__ __align__(16) float Bq[256], Bk[256], Bv[256];
  const int t = threadIdx.x;
  const int base = blockIdx.x * 16;
  async_copy_tile_padded(Ax, x + (size_t)base * 64);
  async_copy_contig(Bq, Wq, 256 * 4);
  async_copy_contig(Bk, Wk, 256 * 4);
  async_copy_contig(Bv, Wv, 256 * 4);
  async_wait0();
  __syncthreads();
  const int n = t & 15, kh = t >> 4;
  v8f accq = vzero8(); accq = gemm_tile(Ax, 68, Bq, 16, 16, 0, accq);
  v8f acck = vzero8(); acck = gemm_tile(Ax, 68, Bk, 16, 16, 0, acck);
  for (int j = 0; j < 8; ++j) {
    int node = base + j + 8 * kh;
    q[node * 16 + n]  = accq[j];
    kk[node * 16 + n] = acck[j];
  }
  for (int c = 0; c < 4; ++c) {
    v8f accv = vzero8(); accv = gemm_tile(Ax + c * 16, 68, Bv, 16, 16, 0, accv);
    for (int j = 0; j < 8; ++j) {
      int node = base + j + 8 * kh;
      v[node * 64 + c * 16 + n] = accv[j];
    }
  }
}

// ---------------------------------------------------------------------------
// Edge scores: score[e] = sum_n silu(q[dst]+k[src]+ef[e]@We)[n]*alpha[n]
// 16 edges per block (edges contiguous in memory).
// ---------------------------------------------------------------------------
__global__ void k_score(const float* __restrict__ ef, const float* __restrict__ We,
                        const float* __restrict__ alpha,
                        const float* __restrict__ q, const float* __restrict__ kk,
                        const int* __restrict__ src, const int* __restrict__ dst,
                        float* __restrict__ score) {
  __shared__ __align__(16) float Ax[16 * 68];
  __shared__ __align__(16) float Bw[64 * 16];
  __shared__ float sAl[16];
  const int t = threadIdx.x;
  const int ebase = blockIdx.x * 16;
  async_copy_tile_padded(Ax, ef + (size_t)ebase * 64);
  async_copy_contig(Bw, We, 1024 * 4);
  if (t < 16) sAl[t] = alpha[t];
  async_wait0();
  __syncthreads();
  const int n = t & 15, kh = t >> 4;
  v8f acc = vzero8();
  acc = gemm_tile(Ax, 68, Bw, 16, 64, 0, acc);
  const float al = sAl[n];
  for (int j = 0; j < 8; ++j) {
    int m = j + 8 * kh;
    int e = ebase + m;
    int d = dst[e], s = src[e];
    float val = acc[j] + q[d * 16 + n] + kk[s * 16 + n];
    float contrib = val / (1.f + __expf(-val)) * al;   // silu * alpha
    contrib += __shfl_xor(contrib, 1);
    contrib += __shfl_xor(contrib, 2);
    contrib += __shfl_xor(contrib, 4);
    contrib += __shfl_xor(contrib, 8);
    if (n == 0) score[e] = contrib;
  }
}

// ---------------------------------------------------------------------------
// Per-dst-node softmax + aggregate + Wo + residual.  One block (64 thr)/node.
// node_mode 0: node=blk (knn, epd=30);  1: node=blk*14+1 (a2r CA, epd=14)
// out_mode  0: out row = node;          1: out row = blk
// ---------------------------------------------------------------------------
__global__ void k_agg(const float* x, const float* __restrict__ v,
                      const float* __restrict__ score, const int* __restrict__ src,
                      const float* __restrict__ Wo, float* out,
                      int epd, int node_mode, int out_mode) {
  const int blk = blockIdx.x;
  const int node = node_mode ? blk * 14 + 1 : blk;
  const int ebase = node_mode ? blk * 14 : blk * epd;
  __shared__ float sw[32];
  __shared__ int ssrc[32];
  __shared__ float sagg[64];
  __shared__ float sWo[256];
  const int t = threadIdx.x;
  if (t < epd) { sw[t] = score[ebase + t]; ssrc[t] = src[ebase + t]; }
  for (int i = t; i < 256; i += 64) sWo[i] = Wo[i];
  __syncthreads();
  float mx = -1e30f;
  for (int e = 0; e < epd; ++e) mx = fmaxf(mx, sw[e]);
  float zsum = 0.f;
  for (int e = 0; e < epd; ++e) zsum += __expf(sw[e] - mx);
  const float inv = 1.f / (zsum + 1e-9f);
  const int c = t >> 4, ch = t & 15;
  float agg = 0.f;
  for (int e = 0; e < epd; ++e) {
    float w = __expf(sw[e] - mx) * inv;
    agg += w * v[ssrc[e] * 64 + c * 16 + ch];
  }
  sagg[t] = agg;
  __syncthreads();
  float sum = x[node * 64 + t];
  for (int i = 0; i < 16; ++i) sum += sagg[c * 16 + i] * sWo[i * 16 + ch];
  out[(out_mode ? blk : node) * 64 + t] = sum;
}

// ---------------------------------------------------------------------------
// res_new = so3_linear(concat(agg_ca, res), Wf 48x32, bf).  128 thr/residue.
// ---------------------------------------------------------------------------
__global__ void k_fuse(const float* __restrict__ aggca, const float* __restrict__ res_in,
                       const float* __restrict__ Wf, const float* __restrict__ bf,
                       float* __restrict__ res_out) {
  const int r = blockIdx.x, t = threadIdx.x;
  const int c = t >> 5, o = t & 31;
  __shared__ float sA[4 * 48];
  if (t < 64) sA[(t >> 4) * 48 + (t & 15)] = aggca[r * 64 + t];
  { int c2 = t >> 5, f = t & 31; sA[c2 * 48 + 16 + f] = res_in[r * 128 + t]; }
  __syncthreads();
  float sum = (c == 0) ? bf[o] : 0.f;
  for (int f = 0; f < 48; ++f) sum += sA[c * 48 + f] * Wf[f * 32 + o];
  res_out[r * 128 + t] = sum;
}

// ---------------------------------------------------------------------------
// Edge MLP: h = relu([s_dst|s_src|z] @ W1 + b1) @ W2 + b2.  16 edges/block.
// ---------------------------------------------------------------------------
__global__ void k_edge_mlp(const float* __restrict__ s, const float* __restrict__ z,
                           const int* __restrict__ src, const int* __restrict__ dst,
                           const float* __restrict__ W1, const float* __restrict__ b1,
                           const float* __restrict__ W2, const float* __restrict__ b2,
                           float* __restrict__ h_out) {
  __shared__ float Ax[16 * 132];
  __shared__ __align__(16) float sW1[128 * 32];
  __shared__ __align__(16) float sW2[32 * 32];
  __shared__ float sH[16 * 36];
  __shared__ float sb1[32], sb2[32];
  const int t = threadIdx.x;
  const int ebase = blockIdx.x * 16;
  async_copy_contig(sW1, W1, 128 * 32 * 4);
  async_copy_contig(sW2, W2, 32 * 32 * 4);
  for (int m = 0; m < 16; ++m) {
    int e = ebase + m;
    Ax[m * 132 + t]      = s[dst[e] * 128 + t];
    Ax[m * 132 + 32 + t] = s[src[e] * 128 + t];
    Ax[m * 132 + 64 + t] = z[e * 64 + t];
    Ax[m * 132 + 96 + t] = z[e * 64 + 32 + t];
  }
  sb1[t] = b1[t]; sb2[t] = b2[t];
  async_wait0();
  __syncthreads();
  const int n = t & 15, kh = t >> 4;
  for (int nt = 0; nt < 2; ++nt) {
    v8f acc = vzero8();
    acc = gemm_tile(Ax, 132, sW1, 32, 128, nt * 16, acc);
    for (int j = 0; j < 8; ++j) {
      int m = j + 8 * kh;
      sH[m * 36 + nt * 16 + n] = fmaxf(acc[j] + sb1[nt * 16 + n], 0.f);
    }
  }
  __syncthreads();
  for (int nt = 0; nt < 2; ++nt) {
    v8f acc = vzero8();
    acc = gemm_tile(sH, 36, sW2, 32, 32, nt * 16, acc);
    for (int j = 0; j < 8; ++j) {
      int m = j + 8 * kh;
      h_out[(ebase + m) * 32 + nt * 16 + n] = acc[j] + sb2[nt * 16 + n];
    }
  }
}

// ---------------------------------------------------------------------------
// Node update: mean-agg, LN, transition MLP, LN.  32 thr/residue, in-place s.
// ---------------------------------------------------------------------------
__global__ void k_node_update(float* res, const float* __restrict__ h_edge,
                              const float* __restrict__ mask,
                              const float* __restrict__ g1, const float* __restrict__ be1,
                              const float* __restrict__ Wt1, const float* __restrict__ bt1,
                              const float* __restrict__ Wt2, const float* __restrict__ bt2,
                              const float* __restrict__ g2, const float* __restrict__ be2) {
  const int r = blockIdx.x, t = threadIdx.x;
  __shared__ float sb[32], sb2[32];
  float upd = 0.f;
  for (int e = 0; e < KK; ++e) upd += h_edge[(r * KK + e) * 32 + t];
  float sv = res[r * 128 + t] + upd * (1.f / (float)KK) * mask[r];
  sb[t] = sv; __syncthreads();
  float mu = 0.f; for (int f = 0; f < 32; ++f) mu += sb[f]; mu *= (1.f / 32.f);
  float var = 0.f; for (int f = 0; f < 32; ++f) { float d = sb[f] - mu; var += d * d; }
  var *= (1.f / 32.f);
  float s1 = (sv - mu) * rsqrtf(var + 1e-5f) * g1[t] + be1[t];
  __syncthreads(); sb[t] = s1; __syncthreads();
  float acc = bt1[t];
  for (int f = 0; f < 32; ++f) acc += sb[f] * Wt1[f * 32 + t];
  acc = fmaxf(acc, 0.f);
  sb2[t] = acc; __syncthreads();
  float acc2 = bt2[t];
  for (int f = 0; f < 32; ++f) acc2 += sb2[f] * Wt2[f * 32 + t];
  float s2 = s1 + acc2;
  __syncthreads(); sb[t] = s2; __syncthreads();
  float mu2 = 0.f; for (int f = 0; f < 32; ++f) mu2 += sb[f]; mu2 *= (1.f / 32.f);
  float var2 = 0.f; for (int f = 0; f < 32; ++f) { float d = sb[f] - mu2; var2 += d * d; }
  var2 *= (1.f / 32.f);
  res[r * 128 + t] = ((s2 - mu2) * rsqrtf(var2 + 1e-5f) * g2[t] + be2[t]) * mask[r];
}

// ---------------------------------------------------------------------------
// z update: z_new = LN(z + [z|s_dst|s_src] @ Wz + bz, ge, bge). 16 edges/block.
// ---------------------------------------------------------------------------
__global__ void k_zin(const float* __restrict__ s, const float* __restrict__ z_in,
                      const int* __restrict__ src, const int* __restrict__ dst,
                      const float* __restrict__ Wz, const float* __restrict__ bz,
                      const float* __restrict__ ge, const float* __restrict__ bge,
                      float* __restrict__ z_out) {
  __shared__ float Ax[16 * 132];
  __shared__ float sZ[16 * 68];
  const int t = threadIdx.x;
  const int ebase = blockIdx.x * 16;
  for (int m = 0; m < 16; ++m) {
    int e = ebase + m;
    Ax[m * 132 + t]      = z_in[e * 64 + t];
    Ax[m * 132 + 32 + t] = z_in[e * 64 + 32 + t];
    Ax[m * 132 + 64 + t] = s[dst[e] * 128 + t];
    Ax[m * 132 + 96 + t] = s[src[e] * 128 + t];
  }
  __syncthreads();
  const int n = t & 15, kh = t >> 4;
  for (int nt = 0; nt < 4; ++nt) {
    v8f acc = vzero8();
    acc = gemm_tile(Ax, 132, Wz, 64, 128, nt * 16, acc);
    for (int j = 0; j < 8; ++j) {
      int m = j + 8 * kh;
      sZ[m * 68 + nt * 16 + n] = acc[j] + bz[nt * 16 + n];
    }
  }
  __syncthreads();
  if (t < 16) {
    int e = ebase + t;
    float mu = 0.f;
    for (int d2 = 0; d2 < 64; ++d2) mu += z_in[e * 64 + d2] + sZ[t * 68 + d2];
    mu *= (1.f / 64.f);
    float var = 0.f;
    for (int d2 = 0; d2 < 64; ++d2) {
      float dd = z_in[e * 64 + d2] + sZ[t * 68 + d2] - mu; var += dd * dd;
    }
    var *= (1.f / 64.f);
    float rs = rsqrtf(var + 1e-5f);
    for (int d2 = 0; d2 < 64; ++d2) {
      float val = z_in[e * 64 + d2] + sZ[t * 68 + d2];
      z_out[e * 64 + d2] = (val - mu) * rs * ge[d2] + bge[d2];
    }
  }
}

// ---------------------------------------------------------------------------
// r2a impute: impute = so3_linear([atom[ca]|res], Wf 48x16, bf); v_ca = impute@Wv
// ---------------------------------------------------------------------------
__global__ void k_impute(const float* __restrict__ atom_x, const float* __restrict__ res,
                         const int* __restrict__ ca_idx,
                         const float* __restrict__ Wf, const float* __restrict__ bf,
                         const float* __restrict__ Wv,
                         float* __restrict__ impute, float* __restrict__ v_ca) {
  const int r = blockIdx.x, t = threadIdx.x;
  const int c = t >> 4, o = t & 15;
  __shared__ float sA[4 * 48];
  __shared__ float sImp[64];
  const int ca = ca_idx[r];
  sA[c * 48 + o] = atom_x[ca * 64 + t];
  for (int i = t; i < 128; i += 64) {
    int c2 = i >> 5, f = i & 31;
    sA[c2 * 48 + 16 + f] = res[r * 128 + i];
  }
  __syncthreads();
  float sum = (c == 0) ? bf[o] : 0.f;
  for (int f = 0; f < 48; ++f) sum += sA[c * 48 + f] * Wf[f * 16 + o];
  impute[r * 64 + t] = sum;
  sImp[t] = sum;
  __syncthreads();
  float vv = 0.f;
  for (int i = 0; i < 16; ++i) vv += sImp[c * 16 + i] * Wv[i * 16 + o];
  v_ca[r * 64 + t] = vv;
}

// r2a broadcast: single incoming edge per atom -> attn weight = 1/(1+1e-9)
__global__ void k_r2a(float* atom_x, const float* __restrict__ impute,
                      const float* __restrict__ v_ca, const float* __restrict__ Wo) {
  const int i = blockIdx.x, t = threadIdx.x;
  const int c = t >> 4, o = t & 15;
  const int r = i / APR;
  const float attnw = 1.f / (1.f + 1e-9f);
  __shared__ float sv[64];
  sv[t] = v_ca[r * 64 + t] * attnw;
  __syncthreads();
  float base = (i % APR == 1) ? impute[r * 64 + t] : atom_x[i * 64 + t];
  float sum = base;
  for (int ich = 0; ich < 16; ++ich) sum += sv[c * 16 + ich] * Wo[ich * 16 + o];
  atom_x[i * 64 + t] = sum;
}

// ---------------------------------------------------------------------------
// Output head: norm_so3 -> linear -> norm_so3 for mu and logvar.
// ---------------------------------------------------------------------------
__device__ __forceinline__ float norm_so3_val(const float* sx, int t,
                                              const float* g0, const float* b0,
                                              const float* g1) {
  const int c = t >> 5, o = t & 31;
  float mu = 0.f;  for (int f = 0; f < 32; ++f) mu += sx[f];  mu *= (1.f / 32.f);
  float var = 0.f; for (int f = 0; f < 32; ++f) { float d = sx[f] - mu; var += d * d; }
  var *= (1.f / 32.f);
  float rms = 0.f; for (int f = 32; f < 128; ++f) rms += sx[f] * sx[f];
  rms = sqrtf(rms * (1.f / 96.f) + 1e-5f);
  if (c == 0) return (sx[t] - mu) * rsqrtf(var + 1e-5f) * g0[o] + b0[o];
  return sx[t] / rms * g1[o];
}

__global__ void k_out(const float* __restrict__ res,
                      const float* rg0, const float* rb0, const float* rg1,
                      const float* Wmu, const float* bmu,
                      const float* mg0, const float* mb0, const float* mg1,
                      const float* Wlv, const float* blv,
                      const float* lg0, const float* lb0, const float* lg1,
                      float* __restrict__ out) {
  const int r = blockIdx.x, t = threadIdx.x;
  const int c = t >> 5, o = t & 31;
  __shared__ float sA[128], sH[128];
  sA[t] = res[r * 128 + t];
  __syncthreads();
  sH[t] = norm_so3_val(sA, t, rg0, rb0, rg1);
  __syncthreads();
  float a = (c == 0) ? bmu[o] : 0.f;
  for (int f = 0; f < 32; ++f) a += sH[c * 32 + f] * Wmu[f * 32 + o];
  __syncthreads();
  sA[t] = a; __syncthreads();
  out[r * 128 + t] = norm_so3_val(sA, t, mg0, mb0, mg1);
  float lv = (c == 0) ? blv[o] : 0.f;
  for (int f = 0; f < 32; ++f) lv += sH[c * 32 + f] * Wlv[f * 32 + o];
  __syncthreads();
  sA[t] = lv; __syncthreads();
  out[RR * 128 + r * 128 + t] = norm_so3_val(sA, t, lg0, lb0, lg1);
}

// ---------------------------------------------------------------------------
// Host driver
// ---------------------------------------------------------------------------
extern "C" void kernel_launch(void* const* d_in, const int* in_sizes, int n_in,
                              void* d_out, int out_size, void* d_ws, size_t ws_size,
                              hipStream_t stream) {
  (void)in_sizes; (void)n_in; (void)out_size; (void)ws_size;
  auto F = [&](int i) { return (const float*)d_in[i]; };

  const float* in_atom   = F(0);
  const float* in_res    = F(1);
  const float* a2r_feat  = F(2);
  const float* knn_feat  = F(3);
  const float* res_feat  = F(4);
  const float* node_mask = F(5);

  struct Attn { const float *We, *Wk, *Wo, *Wq, *Wv, *alpha; };
  Attn a2a[2] = {{F(6),F(7),F(8),F(9),F(10),F(11)},
                 {F(12),F(13),F(14),F(15),F(16),F(17)}};
  struct Fus { const float* Wf; Attn att; const float* bf; };
  Fus a2r[2] = {{F(18),{F(19),F(20),F(21),F(22),F(23),F(24)},F(25)},
                {F(26),{F(27),F(28),F(29),F(30),F(31),F(32)},F(33)}};
  // out head (sorted-key order): Wlv,Wmu,blv,bmu,lb0,lg0,lg1,mb0,mg0,mg1,rb0,rg0,rg1
  const float *Wlv=F(34),*Wmu=F(35),*blv=F(36),*bmu=F(37),*lb0=F(38),*lg0=F(39),
              *lg1=F(40),*mb0=F(41),*mg0=F(42),*mg1=F(43),*rb0=F(44),*rg0=F(45),*rg1=F(46);
  Fus r2a = {F(47),{F(48),F(49),F(50),F(51),F(52),F(53)},F(54)};
  // r2r: W1,W2,Wt1,Wt2,Wz,b1,b2,be1,be2,bge,bt1,bt2,bz,g1,g2,ge
  struct R2R { const float *W1,*W2,*Wt1,*Wt2,*Wz,*b1,*b2,*be1,*be2,*bge,*bt1,*bt2,*bz,*g1,*g2,*ge; };
  R2R r2r[2] = {{F(55),F(56),F(57),F(58),F(59),F(60),F(61),F(62),F(63),F(64),F(65),F(66),F(67),F(68),F(69),F(70)},
                {F(71),F(72),F(73),F(74),F(75),F(76),F(77),F(78),F(79),F(80),F(81),F(82),F(83),F(84),F(85),F(86)}};
  const int* a2r_ei = (const int*)d_in[87];
  const int* knn_ei = (const int*)d_in[88];
  const int* res_ei = (const int*)d_in[89];
  const int* ca_idx = (const int*)d_in[90];
  const int *a2r_src = a2r_ei, *a2r_dst = a2r_ei + AA;
  const int *knn_src = knn_ei, *knn_dst = knn_ei + EKNN;
  const int *res_src = res_ei, *res_dst = res_ei + ERES;

  // workspace carving
  float* w = (float*)d_ws;
  size_t off = 0;
  auto alloc = [&](size_t n) { float* p = w + off; off += n; return p; };
  float* atom_x = alloc((size_t)AA * 64);
  float* qb     = alloc((size_t)AA * 16);
  float* kb     = alloc((size_t)AA * 16);
  float* vb     = alloc((size_t)AA * 64);
  float* score  = alloc((size_t)EKNN);
  float* aggca  = alloc((size_t)RR * 64);
  float* res_a  = alloc((size_t)RR * 128);
  float* res_b  = alloc((size_t)RR * 128);
  float* h_edge = alloc((size_t)ERES * 32);
  float* z_a    = alloc((size_t)ERES * 64);
  float* z_b    = alloc((size_t)ERES * 64);
  float* imput  = alloc((size_t)RR * 64);
  float* v_ca   = alloc((size_t)RR * 64);

  hipMemcpyAsync(atom_x, in_atom, (size_t)AA * 64 * 4, hipMemcpyDeviceToDevice, stream);
  hipMemcpyAsync(res_a,  in_res,  (size_t)RR * 128 * 4, hipMemcpyDeviceToDevice, stream);
  hipMemcpyAsync(z_a,    res_feat,(size_t)ERES * 64 * 4, hipMemcpyDeviceToDevice, stream);

  float* res_cur = res_a; float* res_nxt = res_b;
  float* z_cur = z_a;     float* z_nxt = z_b;

  for (int l = 0; l < 2; ++l) {
    // --- a2a attention over knn graph (deg 30, sorted dst) ---
    const Attn& pa = a2a[l];
    k_qkv<<<AA / 16, 32, 0, stream>>>(atom_x, pa.Wq, pa.Wk, pa.Wv, qb, kb, vb);
    k_score<<<EKNN / 16, 32, 0, stream>>>(knn_feat, pa.We, pa.alpha, qb, kb,
                                          knn_src, knn_dst, score);
    k_agg<<<AA, 64, 0, stream>>>(atom_x, vb, score, knn_src, pa.Wo, atom_x, KK, 0, 0);

    // --- a2r attention onto CA atoms (deg 14, sorted dst) ---
    const Attn& pt = a2r[l].att;
    k_qkv<<<AA / 16, 32, 0, stream>>>(atom_x, pt.Wq, pt.Wk, pt.Wv, qb, kb, vb);
    k_score<<<AA / 16, 32, 0, stream>>>(a2r_feat, pt.We, pt.alpha, qb, kb,
                                        a2r_src, a2r_dst, score);
    k_agg<<<RR, 64, 0, stream>>>(atom_x, vb, score, a2r_src, pt.Wo, aggca, APR, 1, 1);

    k_fuse<<<RR, 128, 0, stream>>>(aggca, res_cur, a2r[l].Wf, a2r[l].bf, res_nxt);
    { float* tmp = res_cur; res_cur = res_nxt; res_nxt = tmp; }

    // --- res2res ---
    const R2R& pr = r2r[l];
    k_edge_mlp<<<ERES / 16, 32, 0, stream>>>(res_cur, z_cur, res_src, res_dst,
                                             pr.W1, pr.b1, pr.W2, pr.b2, h_edge);
    k_node_update<<<RR, 32, 0, stream>>>(res_cur, h_edge, node_mask,
                                         pr.g1, pr.be1, pr.Wt1, pr.bt1,
                                         pr.Wt2, pr.bt2, pr.g2, pr.be2);
    k_zin<<<ERES / 16, 32, 0, stream>>>(res_cur, z_cur, res_src, res_dst,
                                        pr.Wz, pr.bz, pr.ge, pr.bge, z_nxt);
    { float* tmp = z_cur; z_cur = z_nxt; z_nxt = tmp; }

    // --- r2a broadcast (single-edge softmax == constant weight) ---
    if (l == 0) {
      k_impute<<<RR, 64, 0, stream>>>(atom_x, res_cur, ca_idx, r2a.Wf, r2a.bf,
                                      r2a.att.Wv, imput, v_ca);
      k_r2a<<<AA, 64, 0, stream>>>(atom_x, imput, v_ca, r2a.att.Wo);
    }
  }

  k_out<<<RR, 128, 0, stream>>>(res_cur, rg0, rb0, rg1,
                                Wmu, bmu, mg0, mb0, mg1,
                                Wlv, blv, lg0, lb0, lg1, (float*)d_out);
}